// NeuralODE_72808285601946
// MI455X (gfx1250) — compile-verified
//
#include <hip/hip_runtime.h>
#include <hip/hip_bf16.h>
#include <math.h>

// Latent Neural ODE on MI455X (gfx1250), wave32, f32 WMMA 16x16x4.
//
// Mapping: grid = 256 blocks x 32 threads (1 wave). Each wave owns 16 batch
// rows and runs the whole pipeline (reverse RNN scan -> z0 -> RK4 -> decode)
// sequentially, with every small GEMM as V_WMMA_F32_16X16X4_F32 (fp32, so
// numerics match the reference through the 512+4*511 chained matmul stages).
//
// WMMA register layouts assumed (CDNA5 ISA 7.12.2):
//   A 16x4 f32 : lane m (and m+16) = row m; VGPR0 = {K=0 | K=2}, VGPR1 = {K=1 | K=3}
//   B 4x16 f32 : VGPR0 = {row K=0, lanes 0-15 | row K=2, lanes 16-31},
//                VGPR1 = {row K=1 | row K=3}   (mirrors A's K interleave)
//   C/D 16x16  : VGPR r = rows {r | 8+r}, col = lane & 15
// D -> next-A transposes go through a wave-private LDS tile (stride 33 floats,
// conflict-free column reads); LDS is in-order per wave, no barriers needed.
//
// Activations use the gfx1250 hardware transcendentals (v_tanh_f32 /
// v_exp_f32) branch-free instead of the divergent libm expansions, and the
// per-step global x load is software-prefetched one iteration ahead so its
// latency overlaps the WMMA chain.

#define B_    4096
#define T_    512
#define OBS_  2
#define LAT_  4
#define NH_   20
#define RNNH_ 25
#define ST    33   // LDS row stride in floats (conflict-free for 16-row column reads)

typedef float v2f __attribute__((ext_vector_type(2)));
typedef float v8f __attribute__((ext_vector_type(8)));

__device__ __forceinline__ v8f wmma4(v2f a, v2f b, v8f c) {
  // D = A(16x4) * B(4x16) + C, fp32
  return __builtin_amdgcn_wmma_f32_16x16x4_f32(false, a, false, b, (short)0, c,
                                               false, false);
}

#define LOG2E_F  1.4426950408889634f
#define LOG2E2_F 2.8853900817779268f

__device__ __forceinline__ float fast_exp2(float x) {
#if __has_builtin(__builtin_amdgcn_exp2f)
  return __builtin_amdgcn_exp2f(x);
#else
  return exp2f(x);
#endif
}

__device__ __forceinline__ float fast_rcp(float x) {
#if __has_builtin(__builtin_amdgcn_rcpf)
  return __builtin_amdgcn_rcpf(x);
#else
  return 1.0f / x;
#endif
}

__device__ __forceinline__ float fast_tanh(float x) {
#if __has_builtin(__builtin_amdgcn_tanhf)
  return __builtin_amdgcn_tanhf(x);   // gfx1250 v_tanh_f32
#else
  // branch-free: tanh(x) = (e^{2x}-1)/(e^{2x}+1), clamp avoids inf/inf
  float xc = fminf(fmaxf(x, -9.0f), 9.0f);
  float t  = fast_exp2(xc * LOG2E2_F);
  return (t - 1.0f) * fast_rcp(t + 1.0f);
#endif
}

__device__ __forceinline__ float fast_elu(float x) {
  // x<0 path: exp underflow -> 0 -> elu -> -1 (correct); branch-free select
  float e = fast_exp2(x * LOG2E_F) - 1.0f;
  return (x > 0.0f) ? x : e;
}

__device__ __forceinline__ v8f splat8(float v) {
  v8f c;
#pragma unroll
  for (int r = 0; r < 8; ++r) c[r] = v;
  return c;
}

// Build a B operand tile from row-major W[K][N], zero-padded.
__device__ __forceinline__ v2f make_b(const float* W, int K, int N, int kbase,
                                      int nbase, int lane) {
  int n  = nbase + (lane & 15);
  int k0 = kbase + ((lane < 16) ? 0 : 2);
  v2f r;
  r.x = (k0 < K && n < N) ? W[k0 * N + n] : 0.0f;
  r.y = (k0 + 1 < K && n < N) ? W[(k0 + 1) * N + n] : 0.0f;
  return r;
}

// Per-lane bias value for a C/D tile column (same for all 8 VGPRs).
__device__ __forceinline__ float biasval(const float* b, int nbase, int N,
                                         int lane) {
  int n = nbase + (lane & 15);
  return (n < N) ? b[n] : 0.0f;
}

// Load an A operand (K-pass kbase..kbase+3) from the LDS row-major tile.
__device__ __forceinline__ v2f load_a(const float* buf, int kbase, int lane) {
  int m  = lane & 15;
  int k0 = kbase + ((lane < 16) ? 0 : 2);
  v2f r;
  r.x = buf[m * ST + k0];
  r.y = buf[m * ST + k0 + 1];
  return r;
}

// Store a D tile into the LDS row-major tile at column offset coloff.
__device__ __forceinline__ void store_d(float* buf, v8f d, int coloff,
                                        int nbase, int maxn, int lane) {
  int n    = nbase + (lane & 15);
  int mrow = (lane < 16) ? 0 : 8;
  if (n < maxn) {
#pragma unroll
    for (int r = 0; r < 8; ++r) buf[(mrow + r) * ST + coloff + n] = d[r];
  }
}

__device__ __forceinline__ void tanh8(v8f& v) {
#pragma unroll
  for (int r = 0; r < 8; ++r) v[r] = fast_tanh(v[r]);
}
__device__ __forceinline__ void elu8(v8f& v) {
#pragma unroll
  for (int r = 0; r < 8; ++r) v[r] = fast_elu(v[r]);
}
__device__ __forceinline__ void relu8(v8f& v) {
#pragma unroll
  for (int r = 0; r < 8; ++r) v[r] = fmaxf(v[r], 0.0f);
}

struct FW {  // ODE dynamics f() weights as pre-swizzled B operands + biases
  v2f f1[2];     // 4x20  : 1 K-pass, 2 N-tiles
  v2f f2[5][2];  // 20x20 : 5 K-passes, 2 N-tiles
  v2f f3[5];     // 20x4  : 5 K-passes, 1 N-tile
  float bf1[2], bf2[2], bf3;
};
struct DW {  // decoder weights
  v2f d1[2];   // 4x20
  v2f d2[5];   // 20x2
  float bd1[2], bd2;
};

// One f(z) evaluation; z given/returned in A-layout registers.
__device__ __forceinline__ v2f feval(v2f za, const FW& fw, float* buf,
                                     int lane) {
  // u1 = elu(z @ f1 + b1)
  v8f u0 = splat8(fw.bf1[0]), u1 = splat8(fw.bf1[1]);
  u0 = wmma4(za, fw.f1[0], u0);
  u1 = wmma4(za, fw.f1[1], u1);
  elu8(u0); elu8(u1);
  store_d(buf, u0, 0, 0, NH_, lane);
  store_d(buf, u1, 0, 16, NH_, lane);
  // u2 = elu(u1 @ f2 + b2)
  v8f w0 = splat8(fw.bf2[0]), w1 = splat8(fw.bf2[1]);
#pragma unroll
  for (int p = 0; p < 5; ++p) {
    v2f a = load_a(buf, 4 * p, lane);
    w0 = wmma4(a, fw.f2[p][0], w0);
    w1 = wmma4(a, fw.f2[p][1], w1);
  }
  elu8(w0); elu8(w1);
  store_d(buf, w0, 0, 0, NH_, lane);
  store_d(buf, w1, 0, 16, NH_, lane);
  // o = u2 @ f3 + b3
  v8f o = splat8(fw.bf3);
#pragma unroll
  for (int p = 0; p < 5; ++p) {
    v2f a = load_a(buf, 4 * p, lane);
    o = wmma4(a, fw.f3[p], o);
  }
  // transpose o (D layout, 4 cols) back to A layout via LDS
  store_d(buf, o, 0, 0, LAT_, lane);
  return load_a(buf, 0, lane);
}

// pred_x[b0..b0+15][t] = relu(z @ d1 + b1) @ d2 + b2
__device__ __forceinline__ void decode(v2f za, const DW& dw, float* buf,
                                       int lane, float* out, int b0, int t) {
  v8f g0 = splat8(dw.bd1[0]), g1 = splat8(dw.bd1[1]);
  g0 = wmma4(za, dw.d1[0], g0);
  g1 = wmma4(za, dw.d1[1], g1);
  relu8(g0); relu8(g1);
  store_d(buf, g0, 0, 0, NH_, lane);
  store_d(buf, g1, 0, 16, NH_, lane);
  v8f o = splat8(dw.bd2);
#pragma unroll
  for (int p = 0; p < 5; ++p) {
    v2f a = load_a(buf, 4 * p, lane);
    o = wmma4(a, dw.d2[p], o);
  }
  int n    = lane & 15;
  int mrow = (lane < 16) ? 0 : 8;
  if (n < OBS_) {
#pragma unroll
    for (int r = 0; r < 8; ++r)
      out[((size_t)(b0 + mrow + r) * T_ + t) * OBS_ + n] = o[r];
  }
}

__global__ void __launch_bounds__(32) node_kernel(
    const float* __restrict__ trajs, const float* __restrict__ ts,
    const float* __restrict__ eps, const float* __restrict__ i2h_w,
    const float* __restrict__ i2h_b, const float* __restrict__ h2o_w,
    const float* __restrict__ h2o_b, const float* __restrict__ f1w,
    const float* __restrict__ f1b, const float* __restrict__ f2w,
    const float* __restrict__ f2b, const float* __restrict__ f3w,
    const float* __restrict__ f3b, const float* __restrict__ d1w,
    const float* __restrict__ d1b, const float* __restrict__ d2w,
    const float* __restrict__ d2b, float* __restrict__ out) {
  __shared__ float buf[16 * ST];
  const int lane = threadIdx.x;  // 32 threads = 1 wave
  const int b0   = blockIdx.x * 16;

  // zero LDS tile (h0 = 0; cols 27..32 stay zero forever = K padding)
  for (int i = lane; i < 16 * ST; i += 32) buf[i] = 0.0f;

  // ---------------- Phase 1: reverse RNN scan -----------------------------
  v2f Wb[7][2];
#pragma unroll
  for (int p = 0; p < 7; ++p) {
    Wb[p][0] = make_b(i2h_w, OBS_ + RNNH_, RNNH_, 4 * p, 0, lane);
    Wb[p][1] = make_b(i2h_w, OBS_ + RNNH_, RNNH_, 4 * p, 16, lane);
  }
  const float brnn0 = biasval(i2h_b, 0, RNNH_, lane);
  const float brnn1 = biasval(i2h_b, 16, RNNH_, lane);

  const int xrow = lane >> 1, xc = lane & 1;
  const float* xbase = trajs + (size_t)(b0 + xrow) * T_ * OBS_ + xc;

  float xcur = xbase[(size_t)(T_ - 1) * OBS_];
  for (int t = T_ - 1; t >= 0; --t) {
    buf[xrow * ST + xc] = xcur;  // cols 0,1 = x_t
    // prefetch next step's x so the global latency hides under the WMMAs
    float xnext = (t > 0) ? xbase[(size_t)(t - 1) * OBS_] : 0.0f;
    v8f c0 = splat8(brnn0), c1 = splat8(brnn1);
#pragma unroll
    for (int p = 0; p < 7; ++p) {  // K = 27 (cols 0..26), col 27 zero-padded
      v2f a = load_a(buf, 4 * p, lane);
      c0 = wmma4(a, Wb[p][0], c0);
      c1 = wmma4(a, Wb[p][1], c1);
    }
    tanh8(c0); tanh8(c1);
    store_d(buf, c0, 2, 0, RNNH_, lane);   // h -> cols 2..26
    store_d(buf, c1, 2, 16, RNNH_, lane);
    xcur = xnext;
  }

  // ---------------- Phase 2: h2o head, z0 = eps*exp(.5*lv)+mean -----------
  const int m = lane & 15, half = lane >> 4;
  float acc[4];
#pragma unroll
  for (int j = 0; j < 4; ++j) {
    int n   = half * 4 + j;
    float a = h2o_b[n];
    for (int k = 0; k < RNNH_; ++k) a += buf[m * ST + 2 + k] * h2o_w[k * 8 + n];
    acc[j] = a;
  }
  float mn[4], lv[4], z[4];
#pragma unroll
  for (int j = 0; j < 4; ++j) {
    mn[j] = __shfl(acc[j], m, 32);        // lanes 0-15 computed mean
    lv[j] = __shfl(acc[j], m + 16, 32);   // lanes 16-31 computed logvar
    z[j]  = eps[(size_t)(b0 + m) * LAT_ + j] *
                fast_exp2(0.5f * LOG2E_F * lv[j]) + mn[j];
  }
  const size_t zoff = (size_t)B_ * T_ * OBS_;
  if (lane < 16) {
#pragma unroll
    for (int j = 0; j < 4; ++j) {
      out[zoff + (size_t)(b0 + m) * LAT_ + j] = z[j];
      out[zoff + (size_t)B_ * LAT_ + (size_t)(b0 + m) * LAT_ + j] = mn[j];
    }
  } else {
#pragma unroll
    for (int j = 0; j < 4; ++j)
      out[zoff + (size_t)B_ * 2 * LAT_ + (size_t)(b0 + m) * LAT_ + j] = lv[j];
  }
  v2f za;  // z in A layout
  za.x = (lane < 16) ? z[0] : z[2];
  za.y = (lane < 16) ? z[1] : z[3];

  // ---------------- Phase 3: RK4 + per-step decode ------------------------
  FW fw;
  fw.f1[0] = make_b(f1w, LAT_, NH_, 0, 0, lane);
  fw.f1[1] = make_b(f1w, LAT_, NH_, 0, 16, lane);
#pragma unroll
  for (int p = 0; p < 5; ++p) {
    fw.f2[p][0] = make_b(f2w, NH_, NH_, 4 * p, 0, lane);
    fw.f2[p][1] = make_b(f2w, NH_, NH_, 4 * p, 16, lane);
    fw.f3[p]    = make_b(f3w, NH_, LAT_, 4 * p, 0, lane);
  }
  fw.bf1[0] = biasval(f1b, 0, NH_, lane);
  fw.bf1[1] = biasval(f1b, 16, NH_, lane);
  fw.bf2[0] = biasval(f2b, 0, NH_, lane);
  fw.bf2[1] = biasval(f2b, 16, NH_, lane);
  fw.bf3    = biasval(f3b, 0, LAT_, lane);

  DW dw;
  dw.d1[0] = make_b(d1w, LAT_, NH_, 0, 0, lane);
  dw.d1[1] = make_b(d1w, LAT_, NH_, 0, 16, lane);
#pragma unroll
  for (int p = 0; p < 5; ++p) dw.d2[p] = make_b(d2w, NH_, OBS_, 4 * p, 0, lane);
  dw.bd1[0] = biasval(d1b, 0, NH_, lane);
  dw.bd1[1] = biasval(d1b, 16, NH_, lane);
  dw.bd2    = biasval(d2b, 0, OBS_, lane);

  decode(za, dw, buf, lane, out, b0, 0);  // pred_z[0] = z0

  for (int i = 0; i < T_ - 1; ++i) {
    float dt = ts[i + 1] - ts[i];
    v2f k1 = feval(za, fw, buf, lane);
    v2f zt;
    zt.x = za.x + 0.5f * dt * k1.x;  zt.y = za.y + 0.5f * dt * k1.y;
    v2f k2 = feval(zt, fw, buf, lane);
    zt.x = za.x + 0.5f * dt * k2.x;  zt.y = za.y + 0.5f * dt * k2.y;
    v2f k3 = feval(zt, fw, buf, lane);
    zt.x = za.x + dt * k3.x;         zt.y = za.y + dt * k3.y;
    v2f k4 = feval(zt, fw, buf, lane);
    float s = dt * (1.0f / 6.0f);
    za.x += s * (k1.x + 2.0f * k2.x + 2.0f * k3.x + k4.x);
    za.y += s * (k1.y + 2.0f * k2.y + 2.0f * k3.y + k4.y);
    decode(za, dw, buf, lane, out, b0, i + 1);
  }
}

extern "C" void kernel_launch(void* const* d_in, const int* in_sizes, int n_in,
                              void* d_out, int out_size, void* d_ws,
                              size_t ws_size, hipStream_t stream) {
  (void)in_sizes; (void)n_in; (void)out_size; (void)d_ws; (void)ws_size;
  const float* trajs = (const float*)d_in[0];
  const float* ts    = (const float*)d_in[1];
  const float* eps   = (const float*)d_in[2];
  const float* i2h_w = (const float*)d_in[3];
  const float* i2h_b = (const float*)d_in[4];
  const float* h2o_w = (const float*)d_in[5];
  const float* h2o_b = (const float*)d_in[6];
  const float* f1w   = (const float*)d_in[7];
  const float* f1b   = (const float*)d_in[8];
  const float* f2w   = (const float*)d_in[9];
  const float* f2b   = (const float*)d_in[10];
  const float* f3w   = (const float*)d_in[11];
  const float* f3b   = (const float*)d_in[12];
  const float* d1w   = (const float*)d_in[13];
  const float* d1b   = (const float*)d_in[14];
  const float* d2w   = (const float*)d_in[15];
  const float* d2b   = (const float*)d_in[16];
  float* out = (float*)d_out;

  node_kernel<<<B_ / 16, 32, 0, stream>>>(trajs, ts, eps, i2h_w, i2h_b, h2o_w,
                                          h2o_b, f1w, f1b, f2w, f2b, f3w, f3b,
                                          d1w, d1b, d2w, d2b, out);
}